// NCNPredictor_5231270166653
// MI455X (gfx1250) — compile-verified
//
#include <hip/hip_runtime.h>
#include <hip/hip_bf16.h>

#define NN 10000          // nodes
#define DD 128            // feature dim
#define WPR 313           // ceil(10000/32) bit-words per adjacency row
#define ROWSTRIDE 320     // padded words per packed row (1280 B, aligned)

typedef float v2f __attribute__((ext_vector_type(2)));
typedef float v8f __attribute__((ext_vector_type(8)));

// ---------------------------------------------------------------------------
// Kernel 1: S[k][m] = dot(x[k], Wxs[128*(m+1) : 128*(m+2)]) for m = 0..2,
// via v_wmma_f32_16x16x4_f32. One wave per 16-node tile (625 tiles).
// A: 16x4 tile of x  (lanes 0-15: K=k0,k0+1 ; lanes 16-31: K=k0+2,k0+3)
// B: 4x16, column n = W slice (n+1) for n<3, else 0 (masked by select, no branch)
// D: 16x16 f32 accumulator; VGPR r, lanes 0-15 -> (M=r, N=lane),
//    lanes 16-31 -> (M=r+8, N=lane-16).
// ---------------------------------------------------------------------------
__global__ void ncn_project_s_kernel(const float* __restrict__ x,
                                     const float* __restrict__ Wxs,
                                     float* __restrict__ S) {
    const int tile = blockIdx.x;          // 0..624
    const int lane = threadIdx.x;         // 0..31 (blockDim.x == 32, one wave)
    const int half = lane >> 4;           // 0: K pair {0,1}; 1: K pair {2,3}
    const int l15  = lane & 15;

    const int row = tile * 16 + l15;      // node index for A-matrix row (M)
    const float* __restrict__ xrow = x + (long long)row * DD;

    // B column setup: select, not branch, so EXEC stays all-ones for WMMA.
    const float colmask = (l15 < 3) ? 1.0f : 0.0f;
    const int   colsel  = (l15 < 3) ? (l15 + 1) : 1;   // W slice index (1..3)
    const float* __restrict__ wcol = Wxs + colsel * DD;

    v8f acc = {};
    #pragma unroll 4
    for (int k0 = 0; k0 < DD; k0 += 4) {
        v2f a, b;
        a.x = xrow[k0 + half * 2 + 0];
        a.y = xrow[k0 + half * 2 + 1];
        b.x = wcol[k0 + half * 2 + 0] * colmask;
        b.y = wcol[k0 + half * 2 + 1] * colmask;
        acc = __builtin_amdgcn_wmma_f32_16x16x4_f32(
                  /*neg_a=*/false, a, /*neg_b=*/false, b,
                  /*c_mod=*/(short)0, acc, /*reuse_a=*/false, /*reuse_b=*/false);
    }

    // Store columns 0..2 of the D tile: S[node*4 + col]
    if (l15 < 3) {
        #pragma unroll
        for (int r = 0; r < 8; ++r) {
            const int node = tile * 16 + r + half * 8;
            S[node * 4 + l15] = acc[r];
        }
    }
}

// ---------------------------------------------------------------------------
// Kernel 2: bit-pack the three adjacency matrices. One wave per row;
// lane l tests column w*32+l, __ballot forms the 32-bit word. Fully
// coalesced 128B reads; 1.2 GB streamed exactly once (the roofline term).
// ---------------------------------------------------------------------------
__global__ void ncn_bitpack_kernel(const float* __restrict__ a0,
                                   const float* __restrict__ a1,
                                   const float* __restrict__ a2,
                                   unsigned* __restrict__ bits) {
    const int gw   = (int)((blockIdx.x * blockDim.x + threadIdx.x) >> 5);
    const int lane = threadIdx.x & 31;
    if (gw >= 3 * NN) return;
    const int mat = gw / NN;
    const int row = gw - mat * NN;

    const float* __restrict__ adj = (mat == 0) ? a0 : (mat == 1) ? a1 : a2;
    const float* __restrict__ arow = adj + (long long)row * NN;
    unsigned* __restrict__ brow = bits + ((long long)mat * NN + row) * ROWSTRIDE;

    for (int w = 0; w < WPR; ++w) {
        const int col = w * 32 + lane;
        const float v = (col < NN) ? arow[col] : 0.0f;
        const unsigned long long bal = __ballot(v != 0.0f);   // wave32: low 32 bits
        if (lane == 0) brow[w] = (unsigned)bal;
    }
}

// ---------------------------------------------------------------------------
// Kernel 3: one wave per edge.
//   m01 = row01[i] & row01[j]; m1 = row1[i] & row1[j]; m012 = row012[i] & row012[j]
//   m0 = m01 & ~m1 ; m2 = m012 & ~m01
//   acc += S[k][0] over m0, S[k][1] over m1, S[k][2] over m2  (masks ~empty)
//   plus the fused xij . W[0:128] dot, wave-reduced with shfl_xor (width 32).
// Packed rows (38 MB) are L2-resident after kernel 2.
// ---------------------------------------------------------------------------
__global__ void ncn_edge_kernel(const float* __restrict__ x,
                                const int* __restrict__ tar_ei,
                                const float* __restrict__ Wxs,
                                const float* __restrict__ bxs,
                                const float* __restrict__ S,
                                const unsigned* __restrict__ bits,
                                float* __restrict__ out, int E) {
    const int wave = (int)((blockIdx.x * blockDim.x + threadIdx.x) >> 5);
    const int lane = threadIdx.x & 31;
    if (wave >= E) return;

    const int i = tar_ei[wave];
    const int j = tar_ei[E + wave];

    const unsigned* __restrict__ r01i  = bits + (long long)i * ROWSTRIDE;
    const unsigned* __restrict__ r01j  = bits + (long long)j * ROWSTRIDE;
    const unsigned* __restrict__ r1i   = bits + ((long long)NN + i) * ROWSTRIDE;
    const unsigned* __restrict__ r1j   = bits + ((long long)NN + j) * ROWSTRIDE;
    const unsigned* __restrict__ r012i = bits + ((long long)2 * NN + i) * ROWSTRIDE;
    const unsigned* __restrict__ r012j = bits + ((long long)2 * NN + j) * ROWSTRIDE;

    float acc = 0.0f;

    for (int w = lane; w < WPR; w += 32) {
        const unsigned m01  = r01i[w]  & r01j[w];
        const unsigned m1   = r1i[w]   & r1j[w];
        const unsigned m012 = r012i[w] & r012j[w];
        unsigned m0 = m01  & ~m1;    // cn_0 = cn_0_1 * (1 - cn_1)
        unsigned m2 = m012 & ~m01;   // cn_2 = cn_0_1_2 * (1 - cn_0_1)
        const int base = w * 32;
        unsigned t;
        t = m0; while (t) { int b = __builtin_ctz(t); t &= t - 1; acc += S[(base + b) * 4 + 0]; }
        t = m1; while (t) { int b = __builtin_ctz(t); t &= t - 1; acc += S[(base + b) * 4 + 1]; }
        t = m2; while (t) { int b = __builtin_ctz(t); t &= t - 1; acc += S[(base + b) * 4 + 2]; }
    }

    // xij . W0  (xij = xi * xj elementwise, W slice 0)
    const float* __restrict__ xi = x + (long long)i * DD;
    const float* __restrict__ xj = x + (long long)j * DD;
    #pragma unroll
    for (int d = lane; d < DD; d += 32)
        acc += xi[d] * xj[d] * Wxs[d];

    // wave32 butterfly reduction
    #pragma unroll
    for (int off = 16; off > 0; off >>= 1)
        acc += __shfl_xor(acc, off, 32);

    if (lane == 0) out[wave] = acc + bxs[0];
}

// ---------------------------------------------------------------------------
// Host launcher
// ---------------------------------------------------------------------------
extern "C" void kernel_launch(void* const* d_in, const int* in_sizes, int n_in,
                              void* d_out, int out_size, void* d_ws, size_t ws_size,
                              hipStream_t stream) {
    const float* x      = (const float*)d_in[0];
    const float* adj01  = (const float*)d_in[1];
    const float* adj1   = (const float*)d_in[2];
    const float* adj012 = (const float*)d_in[3];
    const int*   tar    = (const int*)d_in[4];
    const float* Wxs    = (const float*)d_in[5];
    const float* bxs    = (const float*)d_in[6];
    float* out = (float*)d_out;

    const int E = in_sizes[4] / 2;   // 8192

    // Workspace layout:
    //   [0, 160000)                : S, N x 4 floats (cols 0..2 used)
    //   [262144, 262144 + 38.4 MB) : packed bits, 3 matrices x N rows x ROWSTRIDE u32
    char* ws = (char*)d_ws;
    float*    S    = (float*)ws;
    unsigned* bits = (unsigned*)(ws + 262144);

    // 1) Node projections via WMMA: 625 tiles of 16 nodes, one wave each.
    ncn_project_s_kernel<<<NN / 16, 32, 0, stream>>>(x, Wxs, S);

    // 2) Bit-pack adjacencies: 3*N rows, one wave per row, 8 waves/block.
    {
        const int waves = 3 * NN;
        const int threads = waves * 32;
        ncn_bitpack_kernel<<<(threads + 255) / 256, 256, 0, stream>>>(adj01, adj1, adj012, bits);
    }

    // 3) Per-edge fused reduction: one wave per edge, 8 waves/block.
    {
        const int threads = E * 32;
        ncn_edge_kernel<<<(threads + 255) / 256, 256, 0, stream>>>(x, tar, Wxs, bxs, S, bits, out, E);
    }
}